// GaussianRasterizer_5686536700122
// MI455X (gfx1250) — compile-verified
//
#include <hip/hip_runtime.h>
#include <math.h>

#define NG     1024     // N_GAUSS (power of two -> bitonic-sort friendly)
#define IMGH   128
#define IMGW   128
#define NFLD   9        // mx,my,ca,cb,cc,op,r,g,b
#define TILE_W 16
#define TILE_H 8

typedef float v2f __attribute__((ext_vector_type(2)));
typedef float v8f __attribute__((ext_vector_type(8)));

// ---------------------------------------------------------------------------
// Fused rasterize with WMMA exponent evaluation (defined first so its
// CDNA5-specific code leads the disassembly).
//
// Block = 256 threads (8 waves) = one 16x8 pixel tile; each wave owns one
// 16-pixel row. Per 16-Gaussian chunk:
//   power[g, p] = Q[g, 0:6] . Basis[0:6, p]   (tile-centered quadratic form)
// evaluated with two chained V_WMMA_F32_16X16X4_F32 (M=gauss, N=pixel, K=4+2).
// D layout: lane n holds gaussians g0..g0+7 for pixel n, lane n+16 holds
// g0+8..g0+15. Each half-chunk is blended sequentially per lane; the halves
// are composed with the associative alpha-compositing operator
// (C,T) o (C',T') = (C + T*C', T*T') via one xor-16 lane exchange. Sorted
// Gaussian state is staged to LDS with the async global->LDS path (ASYNCcnt).
// ---------------------------------------------------------------------------
__global__ void gs_raster(const float* __restrict__ srt,
                          float* __restrict__ out)
{
    __shared__ float s[NFLD * NG];    // 36 KB

    int tid = threadIdx.x;

    // Stage 36KB: 2304 x B128 async transfers, 9 per lane over 256 lanes.
    // The LDS byte address is taken from the real address of s (low 32 bits
    // of the generic pointer are the LDS offset on gfx1250); passing it into
    // the asm also captures s, so the "memory" clobber is known to cover the
    // LDS writes the TDM-side of the async load performs.
#pragma unroll
    for (int it = 0; it < 9; ++it) {
        int vi = it * 256 + tid;                            // float4 index
        unsigned lds_addr = (unsigned)(size_t)&s[vi * 4];   // LDS byte offset
        const float* g = srt + vi * 4;
        asm volatile("global_load_async_to_lds_b128 %0, %1, off"
                     :: "v"(lds_addr), "v"(g)
                     : "memory");
    }
    asm volatile("s_wait_asynccnt 0" ::: "memory");
    __syncthreads();

    const float* s_mx = s + 0*NG;
    const float* s_my = s + 1*NG;
    const float* s_ca = s + 2*NG;
    const float* s_cb = s + 3*NG;
    const float* s_cc = s + 4*NG;
    const float* s_op = s + 5*NG;
    const float* s_r  = s + 6*NG;
    const float* s_g  = s + 7*NG;
    const float* s_b  = s + 8*NG;

    int  wave = tid >> 5;
    int  lane = tid & 31;
    int  m    = lane & 15;
    bool hi   = (lane >= 16);

    int tx = blockIdx.x % (IMGW / TILE_W);
    int ty = blockIdx.x / (IMGW / TILE_W);
    int px = tx * TILE_W + m;             // this lane's pixel column
    int py = ty * TILE_H + wave;          // this wave's pixel row
    float cx = (float)(tx * TILE_W) + 8.0f;   // tile-center (keeps basis small)
    float cy = (float)(ty * TILE_H) + 4.0f;
    float u  = (float)px + 0.5f - cx;     // in [-7.5, 7.5]
    float v  = (float)py + 0.5f - cy;     // in [-3.5, 3.5]

    // B operand (basis), constant across the gaussian loop.
    // WMMA1 K = {1, u, v, u^2}; WMMA2 K = {v^2, u*v, 0, 0}
    v2f B1, B2;
    B1.x = hi ? v       : 1.0f;    // VGPR0: K0 (lanes<16) / K2 (lanes>=16)
    B1.y = hi ? u * u   : u;       // VGPR1: K1 / K3
    B2.x = hi ? 0.0f    : v * v;   // K4 / K6(=0)
    B2.y = hi ? 0.0f    : u * v;   // K5 / K7(=0)

    float Trun = 1.0f, accr = 0.0f, accg = 0.0f, accb = 0.0f;

    for (int g0 = 0; g0 < NG; g0 += 16) {
        // Per-chunk A operand: quadratic-form coefficients of gaussian g0+m
        // around the tile center.
        int gi = g0 + m;
        float mx = s_mx[gi], my = s_my[gi];
        float ca = s_ca[gi], cb = s_cb[gi], cc = s_cc[gi];
        float ex = mx - cx, ey = my - cy;
        float q0 = -0.5f * (ca*ex*ex + cc*ey*ey) - cb*ex*ey;
        float q1 = ca*ex + cb*ey;
        float q2 = cc*ey + cb*ex;
        float q3 = -0.5f * ca;
        float q4 = -0.5f * cc;
        float q5 = -cb;

        v2f A1, A2;
        A1.x = hi ? q2   : q0;     // VGPR0: K0 / K2
        A1.y = hi ? q3   : q1;     // VGPR1: K1 / K3
        A2.x = hi ? 0.0f : q4;     // K4 / K6
        A2.y = hi ? 0.0f : q5;     // K5 / K7

        // power[g, p] via two chained f32 WMMAs (full fp32 accumulate).
        v8f acc = {};
        acc = __builtin_amdgcn_wmma_f32_16x16x4_f32(
                  false, A1, false, B1, (short)0, acc, false, false);
        acc = __builtin_amdgcn_wmma_f32_16x16x4_f32(
                  false, A2, false, B2, (short)0, acc, false, false);

        // Ordered blend over this lane's 8 gaussians.
        int gbase = g0 + (hi ? 8 : 0);
        float t = 1.0f, rc = 0.0f, gc = 0.0f, bc = 0.0f;
#pragma unroll
        for (int r = 0; r < 8; ++r) {
            int gg = gbase + r;
            float power = fminf(acc[r], 0.0f);
            float Gv    = expf(power);
            float alpha = fminf(s_op[gg] * Gv, 0.99f);
            alpha = (alpha >= (1.0f / 255.0f)) ? alpha : 0.0f;
            float w = alpha * t;               // alpha * T_excl (chunk-local)
            rc = fmaf(w, s_r[gg], rc);
            gc = fmaf(w, s_g[gg], gc);
            bc = fmaf(w, s_b[gg], bc);
            t *= (1.0f - alpha);
        }

        // Compose front (g0..g0+7) then back (g0+8..g0+15).
        float t_p = __shfl_xor(t,  16, 32);
        float r_p = __shfl_xor(rc, 16, 32);
        float g_p = __shfl_xor(gc, 16, 32);
        float b_p = __shfl_xor(bc, 16, 32);
        float tf = hi ? t_p : t;                       // front transmittance
        float rf = hi ? r_p : rc,  rb = hi ? rc : r_p; // front / back colors
        float gf = hi ? g_p : gc,  gb = hi ? gc : g_p;
        float bf = hi ? b_p : bc,  bb = hi ? bc : b_p;
        float Rc = fmaf(tf, rb, rf);
        float Gc = fmaf(tf, gb, gf);
        float Bc = fmaf(tf, bb, bf);
        float Tc = t * t_p;

        accr = fmaf(Trun, Rc, accr);
        accg = fmaf(Trun, Gc, accg);
        accb = fmaf(Trun, Bc, accb);
        Trun *= Tc;
    }

    // Lanes n and n+16 hold identical results for pixel n; low half writes.
    if (!hi) {
        int o = (py * IMGW + px) * 3;
        out[o + 0] = accr;
        out[o + 1] = accg;
        out[o + 2] = accb;
    }
}

// ---------------------------------------------------------------------------
// Kernel 1: per-Gaussian projection.
// ---------------------------------------------------------------------------
__global__ void gs_proj(const float* __restrict__ pos,
                        const float* __restrict__ cov,
                        const float* __restrict__ col,
                        const float* __restrict__ opa,
                        const float* __restrict__ cam,
                        const int*   __restrict__ pH,
                        const int*   __restrict__ pW,
                        float* __restrict__ zkey,
                        float* __restrict__ prm,
                        int N)
{
    int i = blockIdx.x * blockDim.x + threadIdx.x;
    if (i >= N) return;

    float W = (float)pW[0];
    float H = (float)pH[0];
    float focal = 0.5f * W / 0.5773502691896258f;   // tan(pi/6)

    float R00 = cam[0], R01 = cam[1],  R02 = cam[2],  t0 = cam[3];
    float R10 = cam[4], R11 = cam[5],  R12 = cam[6],  t1 = cam[7];
    float R20 = cam[8], R21 = cam[9],  R22 = cam[10], t2 = cam[11];

    float px = pos[3*i+0], py = pos[3*i+1], pz = pos[3*i+2];
    float x = R00*px + R01*py + R02*pz + t0;
    float y = R10*px + R11*py + R12*pz + t1;
    float z = R20*px + R21*py + R22*pz + t2;

    float zc   = fmaxf(z, 1e-6f);
    float invz = 1.0f / zc;
    float mx = focal * x * invz + 0.5f * W;
    float my = focal * y * invz + 0.5f * H;

    // J = [[f/z, 0, -f*x/z^2], [0, f/z, -f*y/z^2]]
    float J00 = focal * invz;
    float J02 = -focal * x * invz * invz;
    float J11 = focal * invz;
    float J12 = -focal * y * invz * invz;

    // symmetrized covariance
    const float* C = cov + 9*i;
    float c00 = C[0], c11 = C[4], c22 = C[8];
    float c01 = 0.5f * (C[1] + C[3]);
    float c02 = 0.5f * (C[2] + C[6]);
    float c12 = 0.5f * (C[5] + C[7]);

    // M = R * Cs
    float M00 = R00*c00 + R01*c01 + R02*c02;
    float M01 = R00*c01 + R01*c11 + R02*c12;
    float M02 = R00*c02 + R01*c12 + R02*c22;
    float M10 = R10*c00 + R11*c01 + R12*c02;
    float M11 = R10*c01 + R11*c11 + R12*c12;
    float M12 = R10*c02 + R11*c12 + R12*c22;
    float M20 = R20*c00 + R21*c01 + R22*c02;
    float M21 = R20*c01 + R21*c11 + R22*c12;
    float M22 = R20*c02 + R21*c12 + R22*c22;

    // S = cov_cam = M * R^T (symmetric)
    float S00 = M00*R00 + M01*R01 + M02*R02;
    float S01 = M00*R10 + M01*R11 + M02*R12;
    float S02 = M00*R20 + M01*R21 + M02*R22;
    float S11 = M10*R10 + M11*R11 + M12*R12;
    float S12 = M10*R20 + M11*R21 + M12*R22;
    float S22 = M20*R20 + M21*R21 + M22*R22;

    // V = J * S  (2x3)
    float V00 = J00*S00 + J02*S02;
    float V01 = J00*S01 + J02*S12;
    float V02 = J00*S02 + J02*S22;
    float V11 = J11*S11 + J12*S12;
    float V12 = J11*S12 + J12*S22;

    // cov2d = V * J^T + 0.3*I
    float a = V00*J00 + V02*J02 + 0.3f;
    float b = V01*J11 + V02*J12;
    float c = V11*J11 + V12*J12 + 0.3f;

    float det      = a*c - b*b;
    float det_safe = (det > 0.0f) ? det : 1.0f;
    float inv_det  = 1.0f / det_safe;
    float ca =  c * inv_det;
    float cb = -b * inv_det;
    float cc =  a * inv_det;
    bool valid = (z > 0.2f) && (det > 0.0f);

    zkey[i]        = z;
    prm[0*NG + i]  = mx;
    prm[1*NG + i]  = my;
    prm[2*NG + i]  = ca;
    prm[3*NG + i]  = cb;
    prm[4*NG + i]  = cc;
    prm[5*NG + i]  = valid ? opa[i] : 0.0f;   // invalid => alpha==0, matches reference
    prm[6*NG + i]  = col[3*i+0];
    prm[7*NG + i]  = col[3*i+1];
    prm[8*NG + i]  = col[3*i+2];
}

// ---------------------------------------------------------------------------
// Kernel 2: single-block LDS bitonic sort on (z, idx) + gather into
// field-major packed array (contiguous 36KB, linear async-stage friendly).
// Stage loops kept rolled: 55 unrolled compare-exchange stages would be
// ~1700 I-cache-resident instructions for a latency-bound single block.
// ---------------------------------------------------------------------------
__global__ void gs_sort_gather(const float* __restrict__ zkey,
                               const float* __restrict__ prm,
                               float* __restrict__ srt,
                               int N)
{
    __shared__ float skey[NG];
    __shared__ int   sidx[NG];
    int i = threadIdx.x;

    skey[i] = (i < N) ? zkey[i] : 3.0e38f;
    sidx[i] = (i < N) ? i : -1;
    __syncthreads();

#pragma clang loop unroll(disable)
    for (int k = 2; k <= NG; k <<= 1) {
#pragma clang loop unroll(disable)
        for (int j = k >> 1; j > 0; j >>= 1) {
            int partner = i ^ j;
            if (partner > i) {
                bool up = ((i & k) == 0);
                float ki = skey[i], kp = skey[partner];
                bool doswap = up ? (ki > kp) : (ki < kp);
                if (doswap) {
                    skey[i] = kp; skey[partner] = ki;
                    int t = sidx[i]; sidx[i] = sidx[partner]; sidx[partner] = t;
                }
            }
            __syncthreads();
        }
    }

    int idx = sidx[i];
#pragma unroll
    for (int f = 0; f < NFLD; ++f)
        srt[f*NG + i] = (idx >= 0) ? prm[f*NG + idx] : 0.0f;
}

// ---------------------------------------------------------------------------
extern "C" void kernel_launch(void* const* d_in, const int* in_sizes, int n_in,
                              void* d_out, int out_size, void* d_ws, size_t ws_size,
                              hipStream_t stream)
{
    const float* pos = (const float*)d_in[0];
    const float* cov = (const float*)d_in[1];
    const float* col = (const float*)d_in[2];
    const float* opa = (const float*)d_in[3];
    const float* cam = (const float*)d_in[4];
    const int*   pH  = (const int*)d_in[5];
    const int*   pW  = (const int*)d_in[6];

    int N = in_sizes[0] / 3;   // 1024

    float* ws   = (float*)d_ws;
    float* zkey = ws;                       // [NG]
    float* prm  = ws + NG;                  // [9*NG] field-major, unsorted
    float* srt  = ws + NG + NFLD * NG;      // [9*NG] field-major, depth-sorted

    float* out = (float*)d_out;

    gs_proj<<<(N + 255) / 256, 256, 0, stream>>>(pos, cov, col, opa, cam,
                                                 pH, pW, zkey, prm, N);
    gs_sort_gather<<<1, NG, 0, stream>>>(zkey, prm, srt, N);
    gs_raster<<<(IMGW / TILE_W) * (IMGH / TILE_H), 256, 0, stream>>>(srt, out);
}